// Model_60533269069824
// MI455X (gfx1250) — compile-verified
//
#include <hip/hip_runtime.h>
#include <stdint.h>

// ---------------------------------------------------------------------------
// Scatter rows (nope||rope) into paged KV buffer, returning full buffer.
//   kv_buffer : [524288, 576] f32   (input image of the buffer)
//   loc       : [32768] i32         (unique slot indices)
//   cache_k_nope : [32768, 512] f32
//   cache_k_rope : [32768, 64]  f32
//   d_out     : [524288, 576] f32   (= kv_buffer with rows at loc replaced)
//
// Pure memory-bandwidth problem (~2.57 GB traffic -> ~110us at 23.3 TB/s).
// Kernel 1 streams kv_buffer -> d_out via the CDNA5 async global<->LDS DMA
// path (ASYNCcnt), kernel 2 scatters the fresh rows with NT b128 vmem ops.
// Stream ordering between the two launches makes the overwrite race-free.
// ---------------------------------------------------------------------------

typedef float v4f __attribute__((ext_vector_type(4)));

#define BUF_SLOTS 524288
#define N_LOC     32768
#define NOPE_DIM  512
#define ROPE_DIM  64
#define ROW_DIM   (NOPE_DIM + ROPE_DIM)   // 576 floats per row
#define ROW_V4    (ROW_DIM / 4)           // 144 float4 per row
#define NOPE_V4   (NOPE_DIM / 4)          // 128
#define ROPE_V4   (ROPE_DIM / 4)          // 16

// ---- bulk copy config (async LDS staging) ----
#define CPY_BLOCK        256
#define WAVES_PER_BLOCK  (CPY_BLOCK / 32)               // 8 waves (wave32)
#define CHUNK_BYTES      512                            // 32 lanes x 16B
#define K_CHUNKS         16                             // async ops in flight per wave
#define STAGE_BYTES      (K_CHUNKS * CHUNK_BYTES)       // 8 KB LDS per wave
#define TOTAL_BYTES      (1207959552u)                  // 524288*576*4
#define TOTAL_GROUPS     (TOTAL_BYTES / STAGE_BYTES)    // 147456
#define CPY_GRID         4608                           // 36864 waves -> 4 groups each

__global__ __launch_bounds__(CPY_BLOCK)
void bulk_copy_async(const float* __restrict__ src, float* __restrict__ dst) {
    __shared__ __align__(16) unsigned char stage[WAVES_PER_BLOCK * STAGE_BYTES];

    const unsigned lane = threadIdx.x & 31u;
    const unsigned wave = threadIdx.x >> 5;
    // 32-bit LDS byte address for this lane's 16B slot in this wave's stage
    const unsigned lds0 =
        (unsigned)(size_t)(&stage[wave * STAGE_BYTES]) + lane * 16u;

    const unsigned gw = blockIdx.x * WAVES_PER_BLOCK + wave;
    const unsigned nW = CPY_GRID * WAVES_PER_BLOCK;

    for (unsigned g = gw; g < TOTAL_GROUPS; g += nW) {
        const unsigned base = g * (unsigned)STAGE_BYTES + lane * 16u;

        // Issue K_CHUNKS async DMA loads (global -> LDS), 512B per wave each.
#pragma unroll
        for (int k = 0; k < K_CHUNKS; ++k) {
            unsigned goff = base + (unsigned)(k * CHUNK_BYTES);
            unsigned la   = lds0 + (unsigned)(k * CHUNK_BYTES);
            asm volatile("global_load_async_to_lds_b128 %0, %1, %2 th:TH_LOAD_NT"
                         :: "v"(la), "v"(goff), "s"(src) : "memory");
        }
        asm volatile("s_wait_asynccnt 0x0" ::: "memory");

        // Issue K_CHUNKS async DMA stores (LDS -> global).
#pragma unroll
        for (int k = 0; k < K_CHUNKS; ++k) {
            unsigned goff = base + (unsigned)(k * CHUNK_BYTES);
            unsigned la   = lds0 + (unsigned)(k * CHUNK_BYTES);
            asm volatile("global_store_async_from_lds_b128 %0, %1, %2 th:TH_STORE_NT"
                         :: "v"(goff), "v"(la), "s"(dst) : "memory");
        }
        asm volatile("s_wait_asynccnt 0x0" ::: "memory");
    }
}

// ---- scatter config ----
#define SC_BLOCK 256
#define SC_TOTAL (N_LOC * ROW_V4)   // 4,718,592 float4 elements

__global__ __launch_bounds__(SC_BLOCK)
void scatter_rows(const int* __restrict__ loc,
                  const v4f* __restrict__ nope,
                  const v4f* __restrict__ rope,
                  v4f* __restrict__ out) {
    unsigned idx = blockIdx.x * SC_BLOCK + threadIdx.x;
    if (idx >= (unsigned)SC_TOTAL) return;

    unsigned row = idx / ROW_V4;
    unsigned col = idx - row * ROW_V4;

    v4f v;
    if (col < NOPE_V4) {
        v = __builtin_nontemporal_load(&nope[row * NOPE_V4 + col]);
    } else {
        v = __builtin_nontemporal_load(&rope[row * ROPE_V4 + (col - NOPE_V4)]);
    }

    unsigned long long slot = (unsigned long long)(unsigned)loc[row];
    __builtin_nontemporal_store(v, &out[slot * ROW_V4 + col]);
}

extern "C" void kernel_launch(void* const* d_in, const int* in_sizes, int n_in,
                              void* d_out, int out_size, void* d_ws, size_t ws_size,
                              hipStream_t stream) {
    (void)in_sizes; (void)n_in; (void)out_size; (void)d_ws; (void)ws_size;

    const float* kv   = (const float*)d_in[0];  // [524288, 576] f32
    const int*   loc  = (const int*)  d_in[1];  // [32768] i32
    const float* nope = (const float*)d_in[2];  // [32768, 512] f32
    const float* rope = (const float*)d_in[3];  // [32768, 64] f32
    float*       out  = (float*)d_out;          // [524288, 576] f32

    // 1) Stream the base buffer into the output (async LDS DMA path).
    bulk_copy_async<<<CPY_GRID, CPY_BLOCK, 0, stream>>>(kv, out);

    // 2) Overwrite the scattered rows with concat(nope, rope).
    const unsigned scBlocks = (SC_TOTAL + SC_BLOCK - 1) / SC_BLOCK; // 18432
    scatter_rows<<<scBlocks, SC_BLOCK, 0, stream>>>(
        loc, (const v4f*)nope, (const v4f*)rope, (v4f*)out);
}